// GAM3_70918499992176
// MI455X (gfx1250) — compile-verified
//
#include <hip/hip_runtime.h>

typedef __attribute__((ext_vector_type(2))) float v2f;
typedef __attribute__((ext_vector_type(8))) float v8f;

#define NB  2
#define HH  192
#define WWID 384
#define HWP (HH * WWID)   // 73728

__device__ __forceinline__ int refl(int i, int n) {
    if (i < 0) i = -i;
    if (i >= n) i = 2 * n - 2 - i;
    return i;
}

// ---------------------------------------------------------------------------
// prep: pack W_qkv [96,64], fused bias[96], folded BN scale/shift params
// bnp layout: s1[0..17] t1[18..35] s2[36..51] t2[52..67] s3[68..83] t3[84..99]
// ---------------------------------------------------------------------------
__global__ void prep_kernel(const float* __restrict__ wq, const float* __restrict__ bq,
                            const float* __restrict__ wk, const float* __restrict__ bk,
                            const float* __restrict__ wv, const float* __restrict__ bv,
                            const float* __restrict__ bn1g, const float* __restrict__ bn1b,
                            const float* __restrict__ bn1m, const float* __restrict__ bn1v,
                            const float* __restrict__ bn2g, const float* __restrict__ bn2b,
                            const float* __restrict__ bn2m, const float* __restrict__ bn2v,
                            const float* __restrict__ bn3g, const float* __restrict__ bn3b,
                            const float* __restrict__ bn3m, const float* __restrict__ bn3v,
                            float* __restrict__ Wqkv, float* __restrict__ bqkv,
                            float* __restrict__ bnp) {
    int i = blockIdx.x * blockDim.x + threadIdx.x;
    if (i < 96 * 64) {
        int r = i >> 6, c = i & 63;
        float v;
        if (r < 16)      v = wq[r * 64 + c];
        else if (r < 32) v = wk[(r - 16) * 64 + c];
        else             v = wv[(r - 32) * 64 + c];
        Wqkv[i] = v;
    }
    if (i < 96) {
        bqkv[i] = (i < 16) ? bq[i] : (i < 32) ? bk[i - 16] : bv[i - 32];
    }
    const float eps = 1e-5f;
    if (i < 18) {
        float s = bn1g[i] / sqrtf(bn1v[i] + eps);
        bnp[i] = s;
        bnp[18 + i] = bn1b[i] - bn1m[i] * s;
    }
    if (i < 16) {
        float s2 = bn2g[i] / sqrtf(bn2v[i] + eps);
        bnp[36 + i] = s2;
        bnp[52 + i] = bn2b[i] - bn2m[i] * s2;
        float s3 = bn3g[i] / sqrtf(bn3v[i] + eps);
        bnp[68 + i] = s3;
        bnp[84 + i] = bn3b[i] - bn3m[i] * s3;
    }
}

// ---------------------------------------------------------------------------
// WMMA fp32 GEMM: Y[n, M, HWP] = Wm[M, Kc] * X[n, Kc, HWP] + bias
// M = MT*16. One wave computes one 16-pixel column tile, MT row tiles.
// V_WMMA_F32_16X16X4_F32 fragment layout (wave32):
//   A 16x4: lanes 0-15 K={0,1}, lanes 16-31 K={2,3}; row = lane&15
//   B 4x16: vgpr0 = K rows {0,2}, vgpr1 = {1,3}; col = lane&15
//   D 16x16: vgpr j -> rows {j, j+8}; col = lane&15
// ---------------------------------------------------------------------------
template <int MT>
__global__ void gemm1x1_wmma(const float* __restrict__ Wm, const float* __restrict__ bias,
                             const float* __restrict__ X, float* __restrict__ Y, int Kc) {
    __shared__ float sW[96 * 64];  // max weight tile (24 KB)
    const int M = MT * 16;
    const int nW = M * Kc;
    for (int i = threadIdx.x; i < nW; i += blockDim.x) sW[i] = Wm[i];
    __syncthreads();

    const int lane = threadIdx.x & 31;
    const int wv   = threadIdx.x >> 5;
    const int half = lane >> 4;   // 0: lanes 0-15, 1: lanes 16-31
    const int l16  = lane & 15;

    const int tilesPerBatch = HWP / 16;
    int tile = blockIdx.x * (blockDim.x >> 5) + wv;
    int n = tile / tilesPerBatch;
    if (n >= NB) return;
    int p0 = (tile - n * tilesPerBatch) * 16;

    const float* Xb = X + (size_t)n * Kc * HWP;

    v8f acc[MT] = {};

    for (int k0 = 0; k0 < Kc; k0 += 4) {
        int cb = k0 + 2 * half;
        v2f b;
        b.x = Xb[(size_t)cb * HWP + p0 + l16];
        b.y = Xb[(size_t)(cb + 1) * HWP + p0 + l16];
#pragma unroll
        for (int t = 0; t < MT; ++t) {
            int row = t * 16 + l16;
            v2f a;
            a.x = sW[row * Kc + k0 + 2 * half];
            a.y = sW[row * Kc + k0 + 2 * half + 1];
            acc[t] = __builtin_amdgcn_wmma_f32_16x16x4_f32(
                false, a, false, b, (short)0, acc[t], false, false);
        }
    }

    float* Yb = Y + (size_t)n * M * HWP;
#pragma unroll
    for (int t = 0; t < MT; ++t) {
#pragma unroll
        for (int j = 0; j < 8; ++j) {
            int row = t * 16 + j + 8 * half;
            Yb[(size_t)row * HWP + p0 + l16] = acc[t][j] + bias[row];
        }
    }
}

// ---------------------------------------------------------------------------
// w1 attention weights: per pixel, 9 logits through 18->16->1 MLP + softmax.
// ---------------------------------------------------------------------------
__global__ void attn_w1_kernel(const float* __restrict__ qkv, const float* __restrict__ bnp,
                               const float* __restrict__ w1a, const float* __restrict__ w1b,
                               const float* __restrict__ b1b, const float* __restrict__ wp,
                               float* __restrict__ w1out) {
    __shared__ float sw1a[16 * 18];
    __shared__ float sw1b[16];
    __shared__ float sbn[68];  // s1,t1,s2,t2
    for (int i = threadIdx.x; i < 288; i += blockDim.x) sw1a[i] = w1a[i];
    if (threadIdx.x < 16) sw1b[threadIdx.x] = w1b[threadIdx.x];
    for (int i = threadIdx.x; i < 68; i += blockDim.x) sbn[i] = bnp[i];
    __syncthreads();

    int idx = blockIdx.x * blockDim.x + threadIdx.x;
    if (idx >= NB * HWP) return;
    int n = idx / HWP;
    int p = idx - n * HWP;
    int h = p / WWID, w = p - h * WWID;

    const float* x1 = qkv + (size_t)n * 96 * HWP;              // rows 0-15
    const float* x2 = qkv + (size_t)n * 96 * HWP + 16 * HWP;   // rows 16-31

    float c1[16];
#pragma unroll
    for (int c = 0; c < 16; ++c) c1[c] = x1[(size_t)c * HWP + p];

    float wp00 = wp[0], wp01 = wp[1], wp10 = wp[2], wp11 = wp[3];
    float bb = b1b[0];

    float logit[9];
#pragma unroll 1
    for (int k = 0; k < 9; ++k) {
        int di = k / 3 - 1, dj = k % 3 - 1;
        int hn = refl(h + di, HH);
        int wn = refl(w + dj, WWID);
        int pn = hn * WWID + wn;
        float u[18];
#pragma unroll
        for (int c = 0; c < 16; ++c) {
            float f = c1[c] - x2[(size_t)c * HWP + pn];
            u[c] = fmaxf(sbn[c] * f + sbn[18 + c], 0.f);
        }
        // position branch: p_center - p_neighbor = W_p * (loc_c - loc_n)
        float dlw = (float)(w - wn) * (2.0f / (WWID - 1));
        float dlh = (float)(h - hn) * (2.0f / (HH - 1));
        float pd0 = wp00 * dlw + wp01 * dlh;
        float pd1 = wp10 * dlw + wp11 * dlh;
        u[16] = fmaxf(sbn[16] * pd0 + sbn[18 + 16], 0.f);
        u[17] = fmaxf(sbn[17] * pd1 + sbn[18 + 17], 0.f);

        float acc = 0.f;
#pragma unroll 1
        for (int r = 0; r < 16; ++r) {
            float hs = 0.f;
#pragma unroll
            for (int c = 0; c < 18; ++c) hs += sw1a[r * 18 + c] * u[c];
            float vv = fmaxf(sbn[36 + r] * hs + sbn[52 + r], 0.f);
            acc += sw1b[r] * vv;
        }
        logit[k] = acc + bb;
    }

    float mx = logit[0];
#pragma unroll
    for (int k = 1; k < 9; ++k) mx = fmaxf(mx, logit[k]);
    float sum = 0.f;
#pragma unroll
    for (int k = 0; k < 9; ++k) { logit[k] = __expf(logit[k] - mx); sum += logit[k]; }
    float inv = 1.f / sum;
    float* o = w1out + (size_t)n * 9 * HWP + p;
#pragma unroll
    for (int k = 0; k < 9; ++k) o[(size_t)k * HWP] = logit[k] * inv;
}

// ---------------------------------------------------------------------------
// w2 attention weights: 16->1 on gg differences + softmax.
// ---------------------------------------------------------------------------
__global__ void attn_w2_kernel(const float* __restrict__ gg, const float* __restrict__ bnp,
                               const float* __restrict__ w2a, const float* __restrict__ b2a,
                               float* __restrict__ w2out) {
    __shared__ float sw2a[16];
    __shared__ float sbn[32];  // s3, t3
    if (threadIdx.x < 16) sw2a[threadIdx.x] = w2a[threadIdx.x];
    if (threadIdx.x < 32) sbn[threadIdx.x] = bnp[68 + threadIdx.x];
    __syncthreads();

    int idx = blockIdx.x * blockDim.x + threadIdx.x;
    if (idx >= NB * HWP) return;
    int n = idx / HWP;
    int p = idx - n * HWP;
    int h = p / WWID, w = p - h * WWID;

    const float* gb = gg + (size_t)n * 16 * HWP;
    float cg[16];
#pragma unroll
    for (int c = 0; c < 16; ++c) cg[c] = gb[(size_t)c * HWP + p];
    float bb = b2a[0];

    float logit[9];
#pragma unroll 1
    for (int k = 0; k < 9; ++k) {
        int di = k / 3 - 1, dj = k % 3 - 1;
        int pn = refl(h + di, HH) * WWID + refl(w + dj, WWID);
        float acc = 0.f;
#pragma unroll
        for (int c = 0; c < 16; ++c) {
            float f = cg[c] - gb[(size_t)c * HWP + pn];
            acc += sw2a[c] * fmaxf(sbn[c] * f + sbn[16 + c], 0.f);
        }
        logit[k] = acc + bb;
    }

    float mx = logit[0];
#pragma unroll
    for (int k = 1; k < 9; ++k) mx = fmaxf(mx, logit[k]);
    float sum = 0.f;
#pragma unroll
    for (int k = 0; k < 9; ++k) { logit[k] = __expf(logit[k] - mx); sum += logit[k]; }
    float inv = 1.f / sum;
    float* o = w2out + (size_t)n * 9 * HWP + p;
#pragma unroll
    for (int k = 0; k < 9; ++k) o[(size_t)k * HWP] = logit[k] * inv;
}

// ---------------------------------------------------------------------------
// 3x3 weighted aggregation: O[n,c,p] = sum_k W[n,k,p] * V[n,c,nbr_k(p)]
// One thread handles 8 channels of one pixel (reuses weights/indices).
// vBStride allows V rows to live inside the packed qkv buffer.
// ---------------------------------------------------------------------------
__global__ void agg_kernel(const float* __restrict__ V, size_t vBStride,
                           const float* __restrict__ Wt, float* __restrict__ O_, int C) {
    int idx = blockIdx.x * blockDim.x + threadIdx.x;
    int cgroups = C / 8;
    int total = NB * cgroups * HWP;
    if (idx >= total) return;
    int p = idx % HWP;
    int rest = idx / HWP;
    int cg = rest % cgroups;
    int n = rest / cgroups;
    int h = p / WWID, w = p - h * WWID;

    const float* Wn = Wt + (size_t)n * 9 * HWP + p;
    const float* Vb = V + (size_t)n * vBStride + (size_t)(cg * 8) * HWP;

    float acc[8] = {0.f, 0.f, 0.f, 0.f, 0.f, 0.f, 0.f, 0.f};
#pragma unroll
    for (int k = 0; k < 9; ++k) {
        int di = k / 3 - 1, dj = k % 3 - 1;
        int pn = refl(h + di, HH) * WWID + refl(w + dj, WWID);
        float wk_ = Wn[(size_t)k * HWP];
#pragma unroll
        for (int c = 0; c < 8; ++c) acc[c] += wk_ * Vb[(size_t)c * HWP + pn];
    }
    float* Ob = O_ + ((size_t)n * C + (size_t)(cg * 8)) * HWP + p;
#pragma unroll
    for (int c = 0; c < 8; ++c) Ob[(size_t)c * HWP] = acc[c];
}

// ---------------------------------------------------------------------------
extern "C" void kernel_launch(void* const* d_in, const int* in_sizes, int n_in,
                              void* d_out, int out_size, void* d_ws, size_t ws_size,
                              hipStream_t stream) {
    const float* x    = (const float*)d_in[0];
    const float* g    = (const float*)d_in[1];
    const float* w_q  = (const float*)d_in[2];
    const float* b_q  = (const float*)d_in[3];
    const float* w_k  = (const float*)d_in[4];
    const float* b_k  = (const float*)d_in[5];
    const float* w_v  = (const float*)d_in[6];
    const float* b_v  = (const float*)d_in[7];
    const float* w_g  = (const float*)d_in[8];
    const float* b_g  = (const float*)d_in[9];
    const float* w_p  = (const float*)d_in[10];
    // d_in[11] = b_p: cancels in the position subtraction
    const float* bn1g = (const float*)d_in[12];
    const float* bn1b = (const float*)d_in[13];
    const float* bn1m = (const float*)d_in[14];
    const float* bn1v = (const float*)d_in[15];
    const float* w1a  = (const float*)d_in[16];
    const float* bn2g = (const float*)d_in[17];
    const float* bn2b = (const float*)d_in[18];
    const float* bn2m = (const float*)d_in[19];
    const float* bn2v = (const float*)d_in[20];
    const float* w1b  = (const float*)d_in[21];
    const float* b1b  = (const float*)d_in[22];
    const float* bn3g = (const float*)d_in[23];
    const float* bn3b = (const float*)d_in[24];
    const float* bn3m = (const float*)d_in[25];
    const float* bn3v = (const float*)d_in[26];
    const float* w2a  = (const float*)d_in[27];
    const float* b2a  = (const float*)d_in[28];

    float* ws = (float*)d_ws;
    float* Wqkv = ws;                                  // 6144
    float* bqkv = Wqkv + 6144;                         // 96 (pad 128)
    float* bnp  = bqkv + 128;                          // 100 (pad 128)
    float* qkv  = bnp + 128;                           // NB*96*HWP
    float* ggb  = qkv + (size_t)NB * 96 * HWP;         // NB*16*HWP
    float* w1b_ = ggb + (size_t)NB * 16 * HWP;         // NB*9*HWP
    float* w2b_ = w1b_ + (size_t)NB * 9 * HWP;         // NB*9*HWP
    float* out1 = w2b_ + (size_t)NB * 9 * HWP;         // NB*64*HWP
    float* outF = (float*)d_out;

    prep_kernel<<<(96 * 64 + 255) / 256, 256, 0, stream>>>(
        w_q, b_q, w_k, b_k, w_v, b_v,
        bn1g, bn1b, bn1m, bn1v, bn2g, bn2b, bn2m, bn2v, bn3g, bn3b, bn3m, bn3v,
        Wqkv, bqkv, bnp);

    // qkv GEMM: [96,64] x [64, P]; gg GEMM: [16,128] x [128, P]
    int waves = NB * HWP / 16;   // 9216 column tiles
    int blocks = waves / 8;      // 8 waves (256 threads) per block
    gemm1x1_wmma<6><<<blocks, 256, 0, stream>>>(Wqkv, bqkv, x, qkv, 64);
    gemm1x1_wmma<1><<<blocks, 256, 0, stream>>>(w_g, b_g, g, ggb, 128);

    int pixBlocks = (NB * HWP + 255) / 256;  // 576
    attn_w1_kernel<<<pixBlocks, 256, 0, stream>>>(qkv, bnp, w1a, w1b, b1b, w_p, w1b_);
    attn_w2_kernel<<<pixBlocks, 256, 0, stream>>>(ggb, bnp, w2a, b2a, w2b_);

    int aggBlocks = (NB * 8 * HWP + 255) / 256;  // 4608
    // x3 lives in qkv rows 32..95 (batch stride 96*HWP)
    agg_kernel<<<aggBlocks, 256, 0, stream>>>(qkv + (size_t)32 * HWP, (size_t)96 * HWP,
                                              w1b_, out1, 64);
    agg_kernel<<<aggBlocks, 256, 0, stream>>>(out1, (size_t)64 * HWP,
                                              w2b_, outF, 64);
}